// FDS_51058571214977
// MI455X (gfx1250) — compile-verified
//
#include <hip/hip_runtime.h>
#include <hip/hip_bf16.h>
#include <stdint.h>

// FDS smooth: out = (f - m1) * sqrt(clip(v2/v1, .1, 10)) + m2, with v1<=0 and
// invalid-bucket / epoch<START_SMOOTH passthrough.
// Fused form: out = f * s + c with s = sqrt(clip(v2/v1,...)) (or 1), c = m2 - m1*s (or 0).

#define N_BUCKETS    100
#define FDIM         128
#define TAB_ELEMS    (2 * N_BUCKETS * FDIM)   // 25600 floats (s table then c table)
#define TAB_BYTES    (TAB_ELEMS * 4)          // 102400 bytes, fits in 320KB WGP LDS
#define CLIP_MIN_F   0.1f
#define CLIP_MAX_F   10.0f
#define START_SMOOTH 1

typedef float v4f __attribute__((ext_vector_type(4)));   // native vector: OK for nontemporal builtins
typedef int   v4i __attribute__((ext_vector_type(4)));
typedef v4i __attribute__((address_space(1))) global_v4i;
typedef v4i __attribute__((address_space(3))) lds_v4i;

// ---------------- CDNA5 async global->LDS helpers ----------------
__device__ __forceinline__ void async_copy_b128(const float* gsrc, float* ldst) {
#if __has_builtin(__builtin_amdgcn_global_load_async_to_lds_b128)
    __builtin_amdgcn_global_load_async_to_lds_b128(
        (global_v4i*)(uintptr_t)gsrc,
        (lds_v4i*)(uint32_t)(uintptr_t)ldst,
        0, 0);
#else
    // GV mode: vaddr = 64-bit global address pair, lds byte address in vgpr.
    unsigned lds_addr = (unsigned)(uintptr_t)ldst;
    asm volatile("global_load_async_to_lds_b128 %0, %1, off"
                 :: "v"(lds_addr), "v"(gsrc)
                 : "memory");
#endif
}

__device__ __forceinline__ void wait_asynccnt0() {
#if __has_builtin(__builtin_amdgcn_s_wait_asynccnt)
    __builtin_amdgcn_s_wait_asynccnt(0);
#else
    asm volatile("s_wait_asynccnt 0" ::: "memory");
#endif
}

// ---------------- Kernel 1: bake fused (scale, bias) table ----------------
__global__ __launch_bounds__(256) void fds_prep_kernel(
    const float* __restrict__ rm,   // running_mean_last_epoch   [100*128]
    const float* __restrict__ rv,   // running_var_last_epoch    [100*128]
    const float* __restrict__ sm,   // smoothed_mean_last_epoch  [100*128]
    const float* __restrict__ sv,   // smoothed_var_last_epoch   [100*128]
    float* __restrict__ tab_g)      // [25600]: s at [0,12800), c at [12800,25600)
{
    int t = blockIdx.x * 256 + threadIdx.x;
    if (t >= N_BUCKETS * FDIM) return;
    float m1 = rm[t], v1 = rv[t], m2 = sm[t], v2 = sv[t];
    float s, c;
    if (v1 > 0.0f) {
        float factor = fminf(fmaxf(v2 / v1, CLIP_MIN_F), CLIP_MAX_F);
        s = __builtin_sqrtf(factor);
        c = m2 - m1 * s;
    } else {
        s = 1.0f;   // v1<=0 passthrough: f*1 + 0 == f
        c = 0.0f;
    }
    tab_g[t] = s;
    tab_g[N_BUCKETS * FDIM + t] = c;
}

// ---------------- Kernel 2: stream features, gather from LDS ----------------
// One wave per row: lane L handles columns [4L, 4L+3] as v4f.
// Features/outputs are touched exactly once -> non-temporal policy, so the
// 100KB fused table + bucket ids stay resident in WGP$/L2.
__global__ __launch_bounds__(256) void fds_main_kernel(
    const v4f* __restrict__ feat4,   // [nrows * 32]
    const int* __restrict__ bkt,     // [nrows]
    const float* __restrict__ tab_g, // [25600]
    const int* __restrict__ epoch_p, // [1]
    v4f*       __restrict__ out4,    // [nrows * 32]
    int nrows)
{
    extern __shared__ float tab[];      // TAB_BYTES, dynamic LDS

    const int tid = threadIdx.x;

    // Stage the fused table: 6400 x b128 async copies per block.
    for (int i = tid; i < TAB_ELEMS / 4; i += 256) {
        async_copy_b128(tab_g + i * 4, &tab[i * 4]);
    }
    wait_asynccnt0();
    __syncthreads();

    const bool smooth = (*epoch_p >= START_SMOOTH);
    const unsigned lane = (unsigned)tid & 31u;
    const unsigned gw   = (blockIdx.x * 256u + (unsigned)tid) >> 5;
    const unsigned nw   = (gridDim.x * 256u) >> 5;

    for (unsigned row = gw; row < (unsigned)nrows; row += nw) {
        const int      b = bkt[row];
        const unsigned e = row * 32u + lane;     // <= 16M, 32-bit safe
        v4f f = __builtin_nontemporal_load(&feat4[e]);
        v4f o = f;
        if (smooth && (unsigned)b < (unsigned)N_BUCKETS) {
            const v4f s4 = *(const v4f*)&tab[b * FDIM + lane * 4];
            const v4f c4 = *(const v4f*)&tab[N_BUCKETS * FDIM + b * FDIM + lane * 4];
            o.x = fmaf(f.x, s4.x, c4.x);
            o.y = fmaf(f.y, s4.y, c4.y);
            o.z = fmaf(f.z, s4.z, c4.z);
            o.w = fmaf(f.w, s4.w, c4.w);
        }
        __builtin_nontemporal_store(o, &out4[e]);
    }
}

// ---------------- Fallback: fully fused, no workspace / no LDS ----------------
__global__ __launch_bounds__(256) void fds_direct_kernel(
    const v4f* __restrict__ feat4,
    const int* __restrict__ bkt,
    const v4f* __restrict__ rm4,
    const v4f* __restrict__ rv4,
    const v4f* __restrict__ sm4,
    const v4f* __restrict__ sv4,
    const int* __restrict__ epoch_p,
    v4f*       __restrict__ out4,
    int nrows)
{
    const bool smooth = (*epoch_p >= START_SMOOTH);
    const unsigned lane = (unsigned)threadIdx.x & 31u;
    const unsigned gw   = (blockIdx.x * 256u + threadIdx.x) >> 5;
    const unsigned nw   = (gridDim.x * 256u) >> 5;

    for (unsigned row = gw; row < (unsigned)nrows; row += nw) {
        const int      b = bkt[row];
        const unsigned e = row * 32u + lane;
        v4f f = __builtin_nontemporal_load(&feat4[e]);
        v4f o = f;
        if (smooth && (unsigned)b < (unsigned)N_BUCKETS) {
            const int ti = b * (FDIM / 4) + (int)lane;
            v4f m1 = rm4[ti], v1 = rv4[ti], m2 = sm4[ti], v2 = sv4[ti];
            #define FDS_ELEM(k)                                                   \
                if (v1.k > 0.0f) {                                                \
                    float fac = fminf(fmaxf(v2.k / v1.k, CLIP_MIN_F), CLIP_MAX_F);\
                    float s = __builtin_sqrtf(fac);                               \
                    o.k = fmaf(f.k - m1.k, s, m2.k);                              \
                }
            FDS_ELEM(x) FDS_ELEM(y) FDS_ELEM(z) FDS_ELEM(w)
            #undef FDS_ELEM
        }
        __builtin_nontemporal_store(o, &out4[e]);
    }
}

extern "C" void kernel_launch(void* const* d_in, const int* in_sizes, int n_in,
                              void* d_out, int out_size, void* d_ws, size_t ws_size,
                              hipStream_t stream) {
    const float* feat  = (const float*)d_in[0];
    const int*   bkt   = (const int*)  d_in[1];
    const float* rm    = (const float*)d_in[2];
    const float* rv    = (const float*)d_in[3];
    const float* sm    = (const float*)d_in[4];
    const float* sv    = (const float*)d_in[5];
    const int*   epoch = (const int*)  d_in[6];
    float*       out   = (float*)d_out;

    const int nrows = in_sizes[0] / FDIM;   // N

    if (ws_size >= (size_t)TAB_BYTES) {
        float* tab_g = (float*)d_ws;
        fds_prep_kernel<<<(N_BUCKETS * FDIM + 255) / 256, 256, 0, stream>>>(
            rm, rv, sm, sv, tab_g);
        fds_main_kernel<<<1024, 256, TAB_BYTES, stream>>>(
            (const v4f*)feat, bkt, tab_g, epoch, (v4f*)out, nrows);
    } else {
        fds_direct_kernel<<<1024, 256, 0, stream>>>(
            (const v4f*)feat, bkt,
            (const v4f*)rm, (const v4f*)rv,
            (const v4f*)sm, (const v4f*)sv,
            epoch, (v4f*)out, nrows);
    }
}